// NoisyTopkRouter_49091476193823
// MI455X (gfx1250) — compile-verified
//
#include <hip/hip_runtime.h>
#include <math.h>

// ---------------------------------------------------------------------------
// NoisyTopkRouter for MI455X (gfx1250, wave32, WMMA).
//
// logits = x@Wg + bg ; std = softplus(x@Wn + bn) ; noisy = logits + eps*std
// top-8 per token -> masked softmax -> probs [N,64]; indices [N,8].
//
// GEMM path: V_WMMA_F32_16X16X4_F32 (exact fp32 -> top-k selection matches
// the fp32 reference). One wave owns a 16-token tile and all 128 output
// columns (64 gate + 64 noise) = 8 f32 16x16 accumulators.
//
// Weights (Wg||Wn, 128 cols) are staged per-workgroup into LDS in K-chunks
// of 64, transposed [col][k] with padded stride so each B fragment is one
// ds_load_b64 (compiler fuses pairs into ds_load_2addr_b64). The LDS chunk
// is double-buffered: chunk c+1's global loads are issued before the WMMA
// block of chunk c and ds_store'd after it, so L2->LDS latency overlaps the
// matrix math; one workgroup barrier per chunk.
// ---------------------------------------------------------------------------

typedef __attribute__((ext_vector_type(2))) float v2f;
typedef __attribute__((ext_vector_type(4))) float v4f;
typedef __attribute__((ext_vector_type(8))) float v8f;

#define N_TOKENS 16384
#define D_MODEL  2048
#define NEXP     64
#define TOPK     8

#define KC           64            // K-chunk staged in LDS
#define KSTR         (KC + 2)      // padded k-stride (spread banks, keep 8B align)
#define NCHUNK       (D_MODEL / KC)
#define WAVES_PER_WG 8
#define WG_THREADS   (WAVES_PER_WG * 32)
#define TOK_PER_WG   (WAVES_PER_WG * 16)
#define F4_PER_THD   ((KC * 128) / (4 * WG_THREADS))   // 8 float4 per thread/chunk

__global__ __launch_bounds__(WG_THREADS)
void noisy_topk_router_kernel(const float* __restrict__ x,
                              const float* __restrict__ eps,
                              const float* __restrict__ Wg,
                              const float* __restrict__ bg,
                              const float* __restrict__ Wn,
                              const float* __restrict__ bn,
                              float* __restrict__ out)
{
    // Double-buffered weight chunk: 2 x (128 cols x KSTR) floats = 67.5 KB.
    // Buffer 0 is reused after the GEMM as per-wave 16x64 noisy-logit tiles.
    __shared__ float smem[2 * 128 * KSTR];

    const int tid     = threadIdx.x;
    const int wave    = tid >> 5;
    const int lane    = tid & 31;
    const int half    = lane >> 4;        // 0: lanes 0-15, 1: lanes 16-31
    const int l16     = lane & 15;
    const int tokBase = (blockIdx.x * WAVES_PER_WG + wave) * 16;
    const float* xrow = x + (size_t)(tokBase + l16) * D_MODEL;   // this lane's A row

    // staging address split for this thread (same every chunk)
    int s_kl[F4_PER_THD], s_c4[F4_PER_THD];
#pragma unroll
    for (int i = 0; i < F4_PER_THD; ++i) {
        int f = i * WG_THREADS + tid;     // float4 index in [k][col] order
        s_kl[i] = f >> 5;                 // 32 float4 per 128-col row
        s_c4[i] = (f & 31) << 2;          // starting column (0..124)
    }

    v8f accG[4], accN[4];
#pragma unroll
    for (int t = 0; t < 4; ++t) { accG[t] = (v8f)0.0f; accN[t] = (v8f)0.0f; }

    v4f stg[F4_PER_THD];

    // ---- prologue: stage chunk 0 into buffer 0 ----
#pragma unroll
    for (int i = 0; i < F4_PER_THD; ++i) {
        const float* src = (s_c4[i] < 64) ? (Wg + s_kl[i] * NEXP + s_c4[i])
                                          : (Wn + s_kl[i] * NEXP + (s_c4[i] - 64));
        stg[i] = *(const v4f*)src;
    }
#pragma unroll
    for (int i = 0; i < F4_PER_THD; ++i) {
        float* dst = smem + s_c4[i] * KSTR + s_kl[i];
        dst[0 * KSTR] = stg[i].x; dst[1 * KSTR] = stg[i].y;
        dst[2 * KSTR] = stg[i].z; dst[3 * KSTR] = stg[i].w;
    }
    __syncthreads();

    for (int c = 0; c < NCHUNK; ++c) {
        const int kb = c * KC;
        float* cur = smem + (c & 1) * (128 * KSTR);
        float* nxt = smem + ((c + 1) & 1) * (128 * KSTR);

        // issue next chunk's global loads (in flight during WMMA below)
        if (c + 1 < NCHUNK) {
            const int kn = kb + KC;
#pragma unroll
            for (int i = 0; i < F4_PER_THD; ++i) {
                const float* src = (s_c4[i] < 64)
                    ? (Wg + (kn + s_kl[i]) * NEXP + s_c4[i])
                    : (Wn + (kn + s_kl[i]) * NEXP + (s_c4[i] - 64));
                stg[i] = *(const v4f*)src;
            }
            // pull next chunk of x into WGP-level caches too
            __builtin_prefetch(xrow + kn, 0, 3);
        }

        // ---- WMMA block: 16 K-steps of 4, 8 WMMA each ----
        for (int ks = 0; ks < KC; ks += 4) {
            // A 16x4 f32 fragment (ISA 7.12.2): lanes 0-15 hold K={k0,k0+1},
            // lanes 16-31 hold K={k0+2,k0+3}, M = lane%16.
            v2f a = *(const v2f*)(xrow + kb + ks + half * 2);
            const int krl = ks + half * 2;       // local k for B fragment
#pragma unroll
            for (int t = 0; t < 4; ++t) {
                int col = t * 16 + l16;
                v2f bGf = *(const v2f*)(cur + col * KSTR + krl);
                v2f bNf = *(const v2f*)(cur + (col + 64) * KSTR + krl);
                accG[t] = __builtin_amdgcn_wmma_f32_16x16x4_f32(
                    false, a, false, bGf, (short)0, accG[t], false, false);
                accN[t] = __builtin_amdgcn_wmma_f32_16x16x4_f32(
                    false, a, false, bNf, (short)0, accN[t], false, false);
            }
        }

        // transpose-store next chunk into the other LDS buffer
        if (c + 1 < NCHUNK) {
#pragma unroll
            for (int i = 0; i < F4_PER_THD; ++i) {
                float* dst = nxt + s_c4[i] * KSTR + s_kl[i];
                dst[0 * KSTR] = stg[i].x; dst[1 * KSTR] = stg[i].y;
                dst[2 * KSTR] = stg[i].z; dst[3 * KSTR] = stg[i].w;
            }
        }
        __syncthreads();   // stores visible; cur free for chunk c+2
    }

    // ---- epilogue: bias + softplus + noisy combine -> LDS tile ----
    // C/D layout: VGPR j holds row M=j (lanes 0-15) / M=j+8 (lanes 16-31).
    float* tile = smem + wave * (16 * NEXP);
#pragma unroll
    for (int j = 0; j < 8; ++j) {
        int rowLocal = j + half * 8;
        int token    = tokBase + rowLocal;
#pragma unroll
        for (int t = 0; t < 4; ++t) {
            int col = t * 16 + l16;
            float g  = accG[t][j] + bg[col];
            float nv = accN[t][j] + bn[col];
            // softplus via hw transcendentals (v_exp_f32 / v_log_f32)
            float sp = fmaxf(nv, 0.0f) + __logf(1.0f + __expf(-fabsf(nv)));
            tile[rowLocal * NEXP + col] = g + eps[(size_t)token * NEXP + col] * sp;
        }
    }
    __syncthreads();

    // ---- top-8 + masked softmax: lanes 0-15 each own one token ----
    if (lane < 16) {
        int token  = tokBase + lane;
        float* row = tile + lane * NEXP;

        float topv[TOPK];
        int   topi[TOPK];
#pragma unroll
        for (int s = 0; s < TOPK; ++s) {
            float best = -INFINITY; int bi = 0;
#pragma unroll 4
            for (int e = 0; e < NEXP; ++e) {
                float v = row[e];
                if (v > best) { best = v; bi = e; }   // first occurrence on ties
            }
            topv[s] = best; topi[s] = bi;
            row[bi] = -INFINITY;                      // exclude from next pass
        }

        float m = topv[0];                            // sorted descending
        float w[TOPK], sum = 0.0f;
#pragma unroll
        for (int s = 0; s < TOPK; ++s) { w[s] = __expf(topv[s] - m); sum += w[s]; }
        float inv = 1.0f / sum;

        float* orow = out + (size_t)token * NEXP;     // 256B aligned
        v4f z = (v4f)0.0f;
#pragma unroll
        for (int e = 0; e < NEXP; e += 4) *(v4f*)(orow + e) = z;
#pragma unroll
        for (int s = 0; s < TOPK; ++s) orow[topi[s]] = w[s] * inv;  // stores stay in order

        float* oidx = out + (size_t)N_TOKENS * NEXP + (size_t)token * TOPK;
        v4f i0 = { (float)topi[0], (float)topi[1], (float)topi[2], (float)topi[3] };
        v4f i1 = { (float)topi[4], (float)topi[5], (float)topi[6], (float)topi[7] };
        *(v4f*)(oidx + 0) = i0;
        *(v4f*)(oidx + 4) = i1;
    }
}

extern "C" void kernel_launch(void* const* d_in, const int* in_sizes, int n_in,
                              void* d_out, int out_size, void* d_ws, size_t ws_size,
                              hipStream_t stream)
{
    (void)in_sizes; (void)n_in; (void)out_size; (void)d_ws; (void)ws_size;
    const float* x   = (const float*)d_in[0];
    const float* eps = (const float*)d_in[1];
    const float* Wg  = (const float*)d_in[2];
    const float* bg  = (const float*)d_in[3];
    const float* Wn  = (const float*)d_in[4];
    const float* bn  = (const float*)d_in[5];
    float* out = (float*)d_out;

    dim3 grid(N_TOKENS / TOK_PER_WG);   // 16384 / 128 = 128 workgroups
    dim3 block(WG_THREADS);             // 256 threads = 8 waves
    noisy_topk_router_kernel<<<grid, block, 0, stream>>>(x, eps, Wg, bg, Wn, bn, out);
}